// FullyConnectedVQCs_16t4t1_orig_12077448036873
// MI455X (gfx1250) — compile-verified
//
#include <hip/hip_runtime.h>
#include <math.h>

typedef float v2f __attribute__((ext_vector_type(2)));
typedef float v8f __attribute__((ext_vector_type(8)));

#define NCIRC   17
#define USTRIDE 512           // per circuit: 256 floats Ur (transposed) + 256 floats Ui
#define UBYTES  (NCIRC * USTRIDE * 4)   // 34816 B
#define WAVES   4             // 4 waves (tiles) per workgroup
#define MULTF   ((float)(3.14159265358979323846 - 1.1920928955078125e-07))

// ---------------------------------------------------------------------------
// Kernel 1: build the 17 per-circuit 16x16 complex unitaries from theta.
// Each block = 1 circuit; thread `col` evolves identity column `col` through
// 4 layers of Rot gates + CNOT ring. Fully unrolled -> state lives in VGPRs.
// Output layout (transposed for WMMA-B use): UT[t*16 + s] = U[s][t].
// ---------------------------------------------------------------------------
__global__ __launch_bounds__(32) void vqc_build_unitaries(
    const float* __restrict__ theta, float* __restrict__ Ug)
{
    const int c   = blockIdx.x;      // circuit 0..16
    const int col = threadIdx.x;
    if (col >= 16) return;

    // circuit -> (theta group g, block b)
    int g, b;
    if      (c < 4)  { g = 0; b = c;      }
    else if (c < 8)  { g = 3; b = c - 4;  }
    else if (c < 12) { g = 4; b = c - 8;  }
    else if (c < 16) { g = 1; b = c - 12; }
    else             { g = 2; b = 0;      }
    const float* w = theta + (g * 4 + b) * 48;   // [l][q][k], strides 12,3,1

    float xr[16], xi[16];
#pragma unroll
    for (int s = 0; s < 16; ++s) { xr[s] = (s == col) ? 1.f : 0.f; xi[s] = 0.f; }

#pragma unroll
    for (int l = 0; l < 4; ++l) {
        // --- Rot(phi, th, om) = RZ(om) RY(th) RZ(phi) on each qubit ---
#pragma unroll
        for (int q = 0; q < 4; ++q) {
            const float phi = w[(l * 4 + q) * 3 + 0];
            const float th  = w[(l * 4 + q) * 3 + 1];
            const float om  = w[(l * 4 + q) * 3 + 2];
            const float cp = cosf(0.5f * (phi + om)), sp = sinf(0.5f * (phi + om));
            const float cm = cosf(0.5f * (phi - om)), sm = sinf(0.5f * (phi - om));
            const float c2 = cosf(0.5f * th),         s2 = sinf(0.5f * th);
            // U00 = e^{-i(phi+om)/2} c ; U01 = -e^{+i(phi-om)/2} s
            // U10 = e^{-i(phi-om)/2} s ; U11 =  e^{+i(phi+om)/2} c
            const float u00r =  cp * c2, u00i = -sp * c2;
            const float u01r = -cm * s2, u01i = -sm * s2;
            const float u10r =  cm * s2, u10i = -sm * s2;
            const float u11r =  cp * c2, u11i =  sp * c2;
            const int mask = 8 >> q;     // qubit 0 = MSB of 4-bit index
#pragma unroll
            for (int s = 0; s < 16; ++s) {
                if ((s & mask) == 0) {
                    const int s1 = s | mask;
                    const float a0r = xr[s],  a0i = xi[s];
                    const float a1r = xr[s1], a1i = xi[s1];
                    xr[s]  = u00r * a0r - u00i * a0i + u01r * a1r - u01i * a1i;
                    xi[s]  = u00r * a0i + u00i * a0r + u01r * a1i + u01i * a1r;
                    xr[s1] = u10r * a0r - u10i * a0i + u11r * a1r - u11i * a1i;
                    xi[s1] = u10r * a0i + u10i * a0r + u11r * a1i + u11i * a1r;
                }
            }
        }
        // --- CNOT ring, range r = (l % 3) + 1 ---
        const int r = (l % 3) + 1;
#pragma unroll
        for (int q = 0; q < 4; ++q) {
            const int t  = (q + r) & 3;
            const int mc = 8 >> q, mt = 8 >> t;
#pragma unroll
            for (int s = 0; s < 16; ++s) {
                if ((s & mc) && !(s & mt)) {
                    const int s1 = s | mt;
                    float tr = xr[s]; xr[s] = xr[s1]; xr[s1] = tr;
                    float ti = xi[s]; xi[s] = xi[s1]; xi[s1] = ti;
                }
            }
        }
    }

    float* Uc = Ug + c * USTRIDE;
#pragma unroll
    for (int s = 0; s < 16; ++s) {
        Uc[col * 16 + s]       = xr[s];   // UTr[t=col][s]
        Uc[256 + col * 16 + s] = xi[s];   // UTi[t=col][s]
    }
}

// ---------------------------------------------------------------------------
// 16x16x16 fp32 GEMM tile via 4 chained V_WMMA_F32_16X16X4_F32.
// D[i][s] = sum_t E[i][t] * UT[t][s]
//   A (16x4 f32): lanes 0-15 hold K{0,1}, lanes 16-31 hold K{2,3}
//   B (4x16 f32): mirrored; C/D: VGPR r -> M = r + 8*(lane>=16), N = lane&15
// ---------------------------------------------------------------------------
__device__ __forceinline__ v8f wmma_16x16_k16(const float* sE, const float* sUc,
                                              int lane)
{
    const int m  = lane & 15;
    const int hi = lane >> 4;
    v8f acc = {0.f, 0.f, 0.f, 0.f, 0.f, 0.f, 0.f, 0.f};
#pragma unroll
    for (int kb = 0; kb < 4; ++kb) {
        v2f a, bb;
        const int k0 = 4 * kb + 2 * hi;
        a.x  = sE[m * 16 + k0];
        a.y  = sE[m * 16 + k0 + 1];
        bb.x = sUc[(k0)     * 16 + m];
        bb.y = sUc[(k0 + 1) * 16 + m];
        acc = __builtin_amdgcn_wmma_f32_16x16x4_f32(
            /*neg_a=*/false, a, /*neg_b=*/false, bb,
            /*c_mod=*/(short)0, acc, /*reuse_a=*/false, /*reuse_b=*/false);
    }
    return acc;
}

// ---------------------------------------------------------------------------
// Kernel 2: 4 waves per workgroup, each wave = one 16-sample tile running all
// 17 circuits. Unitary table is async-DMA'd to LDS once per workgroup.
// ---------------------------------------------------------------------------
__global__ __launch_bounds__(32 * WAVES) void vqc_main(
    const float* __restrict__ x, const float* __restrict__ Ug,
    float* __restrict__ out, int B)
{
    __shared__ float sU[NCIRC * USTRIDE];      // 34816 B: all 17 unitaries
    __shared__ float sH[WAVES][2][256];        // per-wave feature double buffer
    __shared__ float sE[WAVES][256];           // per-wave embedding tile
    __shared__ float sP[WAVES][256];           // per-wave probabilities

    const int tid  = threadIdx.x;
    const int lane = tid & 31;
    const int wid  = tid >> 5;
    const int i0   = (blockIdx.x * WAVES + wid) * 16;

    // ---- async bulk copy of all unitaries: global -> LDS (ASYNCcnt path) ----
    {
        unsigned lds_addr = (unsigned)(unsigned long long)(&sU[0]) + tid * 16u;
        unsigned goff     = tid * 16u;
#pragma unroll 1
        for (int it = 0; it < UBYTES / (32 * WAVES * 16); ++it) {
            asm volatile("global_load_async_to_lds_b128 %0, %1, %2"
                         :: "v"(lds_addr), "v"(goff), "s"(Ug) : "memory");
            lds_addr += 32 * WAVES * 16;
            goff     += 32 * WAVES * 16;
        }
        asm volatile("s_wait_asynccnt 0x0" ::: "memory");
    }

    if (lane < 16) {
        const int s = i0 + lane;
        float* h = &sH[wid][0][lane * 16];
        h[0] = 0.f;
#pragma unroll
        for (int m = 0; m < 13; ++m) h[1 + m] = (s < B) ? x[s * 13 + m] : 0.f;
        h[14] = 0.f; h[15] = 0.f;
    }
    __syncthreads();

    int cur = 0;
    for (int stage = 0; stage < 5; ++stage) {
        const int nblk = (stage < 4) ? 4 : 1;
        const int nxt  = cur ^ 1;
        for (int blk = 0; blk < nblk; ++blk) {
            const int circ = (stage < 4) ? (stage * 4 + blk) : 16;

            // ---- RY embedding of this block's 4 angles -> E[sample][16] ----
            if (lane < 16) {
                float cq[4], sq[4];
#pragma unroll
                for (int m = 0; m < 4; ++m) {
                    int f;
                    if (stage == 0)      f = 4 * blk + m;  // 'first': contiguous
                    else if (stage == 4) f = m;            // final 4->1
                    else                 f = 4 * m + blk;  // 'multiple': transpose
                    const float a = sH[wid][cur][lane * 16 + f];
                    cq[m] = cosf(0.5f * a);
                    sq[m] = sinf(0.5f * a);
                }
#pragma unroll
                for (int t = 0; t < 16; ++t) {
                    float e = 1.f;
#pragma unroll
                    for (int q = 0; q < 4; ++q)
                        e *= (t & (8 >> q)) ? sq[q] : cq[q];
                    sE[wid][lane * 16 + t] = e;
                }
            }
            __syncthreads();

            // ---- amplitudes: amp = U * e (real & imag GEMMs via WMMA) ----
            const float* Ur = &sU[circ * USTRIDE];
            const float* Ui = Ur + 256;
            v8f ar = wmma_16x16_k16(&sE[wid][0], Ur, lane);
            v8f ai = wmma_16x16_k16(&sE[wid][0], Ui, lane);

            // ---- probabilities |amp|^2 -> P[sample][basis] ----
            const int n  = lane & 15;
            const int hi = lane >> 4;
#pragma unroll
            for (int r = 0; r < 8; ++r) {
                const float pr = ar[r] * ar[r] + ai[r] * ai[r];
                sP[wid][(r + 8 * hi) * 16 + n] = pr;
            }
            __syncthreads();

            // ---- Pauli-Z expvals + scatter into next feature buffer ----
            if (lane < 16) {
                if (stage < 3) {
#pragma unroll
                    for (int q = 0; q < 4; ++q) {
                        float z = 0.f;
#pragma unroll
                        for (int s = 0; s < 16; ++s)
                            z += (s & (8 >> q)) ? -sP[wid][lane * 16 + s]
                                                :  sP[wid][lane * 16 + s];
                        sH[wid][nxt][lane * 16 + 4 * blk + q] = z;
                    }
                } else {
                    float z = 0.f;
#pragma unroll
                    for (int s = 0; s < 16; ++s)
                        z += (s & 8) ? -sP[wid][lane * 16 + s]
                                     :  sP[wid][lane * 16 + s];
                    if (stage == 3) {
                        sH[wid][nxt][lane * 16 + blk] = z;   // 16->4 readout
                    } else {
                        const int si = i0 + lane;            // final 4->1 readout
                        if (si < B) out[si] = z * MULTF;
                    }
                }
            }
            __syncthreads();
        }
        cur = nxt;
    }
}

// ---------------------------------------------------------------------------
extern "C" void kernel_launch(void* const* d_in, const int* in_sizes, int n_in,
                              void* d_out, int out_size, void* d_ws, size_t ws_size,
                              hipStream_t stream)
{
    const float* x     = (const float*)d_in[0];   // (B, 13) fp32
    const float* theta = (const float*)d_in[1];   // (5,4,4,4,3) fp32
    float*       out   = (float*)d_out;           // (B, 1)  fp32
    float*       Ug    = (float*)d_ws;            // 17 * 512 floats = 34816 B

    const int B = in_sizes[0] / 13;

    vqc_build_unitaries<<<NCIRC, 32, 0, stream>>>(theta, Ug);
    const int tilesPerBlock = WAVES * 16;
    vqc_main<<<(B + tilesPerBlock - 1) / tilesPerBlock, 32 * WAVES, 0, stream>>>(
        x, Ug, out, B);
}